// GNN_62654982914513
// MI455X (gfx1250) — compile-verified
//
#include <hip/hip_runtime.h>
#include <hip/hip_bf16.h>

#define N_NODES 50000
#define D_HID   256

typedef __attribute__((ext_vector_type(2))) float v2f;
typedef __attribute__((ext_vector_type(8))) float v8f;

// ---------------------------------------------------------------------------
// Dense GEMM H = X @ W using V_WMMA_F32_16X16X4_F32 (fp32, full precision).
// One wave computes a 16x64 tile: A-fragment reused across 4 accumulators.
// Block = 128 threads (4 waves) -> 16 rows x 256 cols per block.
// DIN/DOUT are compile-time so the fully-unrolled k-loop uses only
// base-pointer + 24-bit immediate addressing (no per-iter 64-bit adds).
// Layouts (wave32):
//   A 16x4 : lane = half*16+m ; vgpr j = A[m][2*half+j]
//   B 4x16 : lane = half*16+n ; vgpr j = B[2*half+j][n]
//   C 16x16: vgpr r -> row r+8*half ; lane -> col n
// ---------------------------------------------------------------------------
template <int DIN>
__launch_bounds__(128)
__global__ void gcn_gemm_wmma(const float* __restrict__ X,
                              const float* __restrict__ W,
                              float* __restrict__ H) {
  constexpr int DOUT = D_HID;

  const int lane = threadIdx.x & 31;
  const int wave = threadIdx.x >> 5;
  const int half = lane >> 4;
  const int m    = lane & 15;          // row index within tile (also col index n)

  const int row0 = blockIdx.x * 16;
  const int col0 = wave * 64;          // 4 waves * 64 cols = 256

  const float* __restrict__ Xrow  = X + (size_t)(row0 + m) * DIN + 2 * half;
  const float* __restrict__ Wbase = W + (size_t)(2 * half) * DOUT + col0 + m;

  v8f acc0 = {0.f,0.f,0.f,0.f,0.f,0.f,0.f,0.f};
  v8f acc1 = acc0, acc2 = acc0, acc3 = acc0;

  #pragma unroll
  for (int k = 0; k < DIN; k += 4) {
    v2f a;
    a.x = Xrow[k];
    a.y = Xrow[k + 1];

    const float* __restrict__ w0 = Wbase + (size_t)k * DOUT;  // row ka
    const float* __restrict__ w1 = w0 + DOUT;                 // row ka+1

    v2f b0, b1, b2, b3;
    b0.x = w0[0];   b0.y = w1[0];
    b1.x = w0[16];  b1.y = w1[16];
    b2.x = w0[32];  b2.y = w1[32];
    b3.x = w0[48];  b3.y = w1[48];

    acc0 = __builtin_amdgcn_wmma_f32_16x16x4_f32(false, a, false, b0, (short)0, acc0, false, false);
    acc1 = __builtin_amdgcn_wmma_f32_16x16x4_f32(false, a, false, b1, (short)0, acc1, false, false);
    acc2 = __builtin_amdgcn_wmma_f32_16x16x4_f32(false, a, false, b2, (short)0, acc2, false, false);
    acc3 = __builtin_amdgcn_wmma_f32_16x16x4_f32(false, a, false, b3, (short)0, acc3, false, false);
  }

  float* __restrict__ Hb = H + (size_t)row0 * DOUT + col0 + m;
  #pragma unroll
  for (int r = 0; r < 8; ++r) {
    float* p = Hb + (size_t)(r + 8 * half) * DOUT;
    p[0]  = acc0[r];
    p[16] = acc1[r];
    p[32] = acc2[r];
    p[48] = acc3[r];
  }
}

// ---------------------------------------------------------------------------
// Degree kernels: dinv starts at 1.0 (self loop), edges atomically add 1,
// then in-place rsqrt. Integer-valued fp adds -> order-independent, exact.
// ---------------------------------------------------------------------------
__global__ void gcn_deg_init(float* __restrict__ dinv, int n) {
  int i = blockIdx.x * blockDim.x + threadIdx.x;
  if (i < n) dinv[i] = 1.0f;
}

__global__ void gcn_deg_edges(const int* __restrict__ dst, float* __restrict__ dinv, int e) {
  int i = blockIdx.x * blockDim.x + threadIdx.x;
  if (i < e) unsafeAtomicAdd(&dinv[dst[i]], 1.0f);
}

__global__ void gcn_deg_rsqrt(float* __restrict__ dinv, int n) {
  int i = blockIdx.x * blockDim.x + threadIdx.x;
  if (i < n) dinv[i] = rsqrtf(dinv[i]);   // deg >= 1 always (self loop)
}

// AGG[i][f] = b[f]  (fused zero + bias)
__global__ void gcn_init_bias(float* __restrict__ agg, const float* __restrict__ b, int total) {
  int i = blockIdx.x * blockDim.x + threadIdx.x;
  if (i < total) agg[i] = b[i & (D_HID - 1)];
}

// ---------------------------------------------------------------------------
// Edge aggregation: one wave per edge. Lane l owns features [8l, 8l+8).
// Reads H[src] (L2-resident), scales by dinv[s]*dinv[d], hardware fp32
// atomics into AGG[dst].
// ---------------------------------------------------------------------------
__launch_bounds__(256)
__global__ void gcn_edge_agg(const int* __restrict__ src, const int* __restrict__ dst,
                             const float* __restrict__ dinv,
                             const float* __restrict__ H,
                             float* __restrict__ agg, int e) {
  int gid  = blockIdx.x * blockDim.x + threadIdx.x;
  int edge = gid >> 5;
  int lane = gid & 31;
  if (edge >= e) return;

  int s = src[edge], d = dst[edge];
  float nrm = dinv[s] * dinv[d];

  const float4* __restrict__ hrow = (const float4*)(H + (size_t)s * D_HID);
  float4 h0 = hrow[lane * 2 + 0];
  float4 h1 = hrow[lane * 2 + 1];

  float* __restrict__ a = agg + (size_t)d * D_HID + lane * 8;
  unsafeAtomicAdd(a + 0, h0.x * nrm);
  unsafeAtomicAdd(a + 1, h0.y * nrm);
  unsafeAtomicAdd(a + 2, h0.z * nrm);
  unsafeAtomicAdd(a + 3, h0.w * nrm);
  unsafeAtomicAdd(a + 4, h1.x * nrm);
  unsafeAtomicAdd(a + 5, h1.y * nrm);
  unsafeAtomicAdd(a + 6, h1.z * nrm);
  unsafeAtomicAdd(a + 7, h1.w * nrm);
}

// out[i][f] = maybe_relu( AGG[i][f] + H[i][f] * dinv[i]^2 )   (self-loop fused)
__global__ void gcn_finalize(const float* __restrict__ agg, const float* __restrict__ H,
                             const float* __restrict__ dinv,
                             float* __restrict__ out, int do_relu, int total) {
  int gid = blockIdx.x * blockDim.x + threadIdx.x;
  if (gid >= total) return;
  int i = gid >> 8;                     // / D_HID
  float c = dinv[i];
  float v = agg[gid] + H[gid] * c * c;
  out[gid] = do_relu ? fmaxf(v, 0.0f) : v;
}

// ---------------------------------------------------------------------------
extern "C" void kernel_launch(void* const* d_in, const int* in_sizes, int n_in,
                              void* d_out, int out_size, void* d_ws, size_t ws_size,
                              hipStream_t stream) {
  const float* x   = (const float*)d_in[0];   // [N, 128]
  const int*   ei  = (const int*)  d_in[1];   // [2, E]
  const float* W1  = (const float*)d_in[2];
  const float* b1  = (const float*)d_in[3];
  const float* W2  = (const float*)d_in[4];
  const float* b2  = (const float*)d_in[5];
  const float* W3  = (const float*)d_in[6];
  const float* b3  = (const float*)d_in[7];

  const int N = in_sizes[0] / 128;            // 50000
  const int E = in_sizes[1] / 2;              // 500000
  const int total = N * D_HID;                // 12.8M

  const int* src = ei;
  const int* dst = ei + E;

  // workspace: H [N,256] | AGG [N,256] | dinv [N]
  float* H    = (float*)d_ws;
  float* AGG  = H + (size_t)total;
  float* dinv = AGG + (size_t)total;

  const int rowTiles = N / 16;                // 3125
  dim3 gemmGrid(rowTiles, 1, 1);
  dim3 gemmBlk(128, 1, 1);
  const int nBlkN  = (N + 255) / 256;
  const int nBlkE  = (E + 255) / 256;
  const int nBlkT  = (total + 255) / 256;
  const int nBlkEW = (E * 32 + 255) / 256;    // one wave per edge

  // ---- degrees ----
  gcn_deg_init <<<nBlkN, 256, 0, stream>>>(dinv, N);
  gcn_deg_edges<<<nBlkE, 256, 0, stream>>>(dst, dinv, E);
  gcn_deg_rsqrt<<<nBlkN, 256, 0, stream>>>(dinv, N);

  // ---- layer 1 (X = input x, relu) ----
  gcn_gemm_wmma<128><<<gemmGrid, gemmBlk, 0, stream>>>(x, W1, H);
  gcn_init_bias<<<nBlkT, 256, 0, stream>>>(AGG, b1, total);
  gcn_edge_agg <<<nBlkEW, 256, 0, stream>>>(src, dst, dinv, H, AGG, E);
  gcn_finalize <<<nBlkT, 256, 0, stream>>>(AGG, H, dinv, AGG, 1, total);   // AGG becomes X2

  // ---- layer 2 (X = AGG, relu) ----
  gcn_gemm_wmma<256><<<gemmGrid, gemmBlk, 0, stream>>>(AGG, W2, H);
  gcn_init_bias<<<nBlkT, 256, 0, stream>>>(AGG, b2, total);                // after GEMM consumed AGG
  gcn_edge_agg <<<nBlkEW, 256, 0, stream>>>(src, dst, dinv, H, AGG, E);
  gcn_finalize <<<nBlkT, 256, 0, stream>>>(AGG, H, dinv, AGG, 1, total);   // AGG becomes X3

  // ---- layer 3 (no relu, write d_out) ----
  gcn_gemm_wmma<256><<<gemmGrid, gemmBlk, 0, stream>>>(AGG, W3, H);
  gcn_init_bias<<<nBlkT, 256, 0, stream>>>(AGG, b3, total);
  gcn_edge_agg <<<nBlkEW, 256, 0, stream>>>(src, dst, dinv, H, AGG, E);
  gcn_finalize <<<nBlkT, 256, 0, stream>>>(AGG, H, dinv, (float*)d_out, 0, total);
}